// Layer_64759516889476
// MI455X (gfx1250) — compile-verified
//
#include <hip/hip_runtime.h>
#include <math.h>

typedef __attribute__((ext_vector_type(2))) float v2f;
typedef __attribute__((ext_vector_type(8))) float v8f;

#define NWAVES 4          // waves per block in node kernel, 1 tile of 16 nodes per wave
#define OUTC   288

__device__ __forceinline__ v8f v8zero() {
  v8f z = {0.f,0.f,0.f,0.f,0.f,0.f,0.f,0.f};
  return z;
}

__device__ __forceinline__ float gelu_tanh(float x) {
  // jax.nn.gelu default (approximate=True)
  float x3 = x * x * x;
  float t = tanhf(0.7978845608028654f * (x + 0.044715f * x3));
  return 0.5f * x * (1.0f + t);
}

__device__ __forceinline__ void atomic_add_f32(float* p, float v) {
  __hip_atomic_fetch_add(p, v, __ATOMIC_RELAXED, __HIP_MEMORY_SCOPE_AGENT);
}

// ---------------------------------------------------------------------------
// WMMA helpers.  f32 16x16x4: A is 16x4 (2 VGPR), B is 4x16 (2 VGPR), C/D 8 VGPR.
// A layout: lanes 0-15 -> M=lane, K=k0+{0,1}; lanes 16-31 -> M=lane-16, K=k0+{2,3}
// B layout (mirror of A): lanes 0-15 -> N=lane, K=k0+{0,1}; lanes 16-31 -> K=k0+{2,3}
// C/D: VGPR r, lanes0-15 => (M=r, N=lane), lanes16-31 => (M=r+8, N=lane-16)
// Weights in LDS pair-interleaved: lw[(k>>1)*64 + col*2 + (k&1)] = W[k][col]
// ---------------------------------------------------------------------------
template<int K>
__device__ __forceinline__ void gemm16(const float* __restrict__ Abase, int lda, int sk,
                                       const float* __restrict__ lwB,
                                       int row, int lane, v8f acc[2]) {
  const int off  = (lane >> 4) << 1;   // 0 or 2
  const int colL = lane & 15;
#pragma unroll
  for (int k0 = 0; k0 < K; k0 += 4) {
    const int kk = k0 + off;           // even
    v2f a;
    a.x = Abase[row * lda + kk * sk];
    a.y = Abase[row * lda + (kk + 1) * sk];
#pragma unroll
    for (int nt = 0; nt < 2; ++nt) {
      const float* bp = lwB + (kk >> 1) * 64 + (colL + nt * 16) * 2;
      v2f b;
      b.x = bp[0];
      b.y = bp[1];
      acc[nt] = __builtin_amdgcn_wmma_f32_16x16x4_f32(
          false, a, false, b, (short)0, acc[nt], false, false);
    }
  }
}

__device__ __forceinline__ void stage16x32(float* st, const v8f acc[2], int lane) {
  const int colL = lane & 15;
  const int mb = (lane >= 16) ? 8 : 0;
#pragma unroll
  for (int nt = 0; nt < 2; ++nt)
#pragma unroll
    for (int r = 0; r < 8; ++r)
      st[(mb + r) * 32 + colL + nt * 16] = acc[nt][r];
}

__device__ __forceinline__ void store_out(float* __restrict__ out, int n0, int N, int lane,
                                          const v8f o[2], int colBase, int colStride, int colOff) {
  const int colL = lane & 15;
  const int mb = (lane >= 16) ? 8 : 0;
#pragma unroll
  for (int nt = 0; nt < 2; ++nt)
#pragma unroll
    for (int r = 0; r < 8; ++r) {
      int n = n0 + mb + r;
      if (n < N) {
        int col = colBase + (colL + nt * 16) * colStride + colOff;
        out[(size_t)n * OUTC + col] = o[nt][r];
      }
    }
}

// ---------------------------------------------------------------------------
__global__ void zero_kernel(float* __restrict__ p, size_t n) {
  size_t i = (size_t)blockIdx.x * blockDim.x + threadIdx.x;
  size_t stride = (size_t)gridDim.x * blockDim.x;
  for (; i < n; i += stride) p[i] = 0.0f;
}

// ---------------------------------------------------------------------------
// Edge kernel: one wave32 per edge; lane = channel (C == 32 == wave size).
// Accumulators (scratch, all /16 folded in, everything L2-resident):
//   a0: [N][64]      (cols 0-31 = x0, 32-63 = tp_0e)
//   a1: [N][3][64]   (i-major so node-phase k-pairs are contiguous)
//   a2: [N][5][32]   (m-major)
// ---------------------------------------------------------------------------
__global__ __launch_bounds__(256) void edge_kernel(
    const float* __restrict__ node0, const float* __restrict__ node1,
    const float* __restrict__ pos,
    const int* __restrict__ senders, const int* __restrict__ receivers,
    float* __restrict__ a0, float* __restrict__ a1, float* __restrict__ a2,
    int E) {
  const int wave = threadIdx.x >> 5;
  const int lane = threadIdx.x & 31;
  const int e = blockIdx.x * 8 + wave;
  if (e >= E) return;

  const int s = senders[e];     // wave-uniform -> scalar loads
  const int r = receivers[e];

  float rx = pos[r * 3 + 0] - pos[s * 3 + 0];
  float ry = pos[r * 3 + 1] - pos[s * 3 + 1];
  float rz = pos[r * 3 + 2] - pos[s * 3 + 2];
  float rn = sqrtf(rx * rx + ry * ry + rz * rz);
  float inv = 1.7320508075688772f / fmaxf(rn, 1e-9f);   // sqrt(3)/|r|
  float sx = rx * inv, sy = ry * inv, sz = rz * inv;

  const int c = lane;
  float x0 = node0[(size_t)s * 32 + c];
  const float* xp = node1 + (size_t)s * 96 + c * 3;
  float vx = xp[0], vy = xp[1], vz = xp[2];

  const float scale = 0.0625f;   // 1/DEN
  float tp0 = (vx * sx + vy * sy + vz * sz) * 0.5773502691896258f;  // /sqrt(3)

  const float c2 = 0.7071067811865476f;   // 1/sqrt(2)
  const float c6 = 0.4082482904638631f;   // 1/sqrt(6)
  float t2[5];
  t2[0] = c2 * (vx * sy + vy * sx);
  t2[1] = c2 * (vy * sz + vz * sy);
  t2[2] = c6 * (2.0f * vz * sz - vx * sx - vy * sy);
  t2[3] = c2 * (vx * sz + vz * sx);
  t2[4] = c2 * (vx * sx - vy * sy);
  // note: the 1e (cross product) branch of the reference is dead code -> skipped

  float* p0 = a0 + (size_t)r * 64;
  atomic_add_f32(p0 + c, x0 * scale);
  atomic_add_f32(p0 + 32 + c, tp0 * scale);

  float sh[3] = {sx, sy, sz};
  float vv[3] = {vx, vy, vz};
#pragma unroll
  for (int i = 0; i < 3; ++i) {
    float* p1 = a1 + ((size_t)r * 3 + i) * 64;
    atomic_add_f32(p1 + c, vv[i] * scale);
    atomic_add_f32(p1 + 32 + c, x0 * sh[i] * scale);
  }
#pragma unroll
  for (int m = 0; m < 5; ++m)
    atomic_add_f32(a2 + ((size_t)r * 5 + m) * 32 + c, t2[m] * scale);
}

// ---------------------------------------------------------------------------
// Node kernel: 1 wave per 16-node tile; all linears as v_wmma_f32_16x16x4_f32.
// Weights pre-staged to LDS pair-interleaved (ds_load_b64 per B fragment).
// Intermediate h tiles staged through per-wave LDS to remap D-layout -> A-layout.
// ---------------------------------------------------------------------------
__global__ __launch_bounds__(NWAVES * 32) void node_kernel(
    const float* __restrict__ a0, const float* __restrict__ a1, const float* __restrict__ a2,
    const float* __restrict__ node0, const float* __restrict__ node1,
    const float* __restrict__ Wpre0, const float* __restrict__ Wpre1, const float* __restrict__ Wpre2,
    const float* __restrict__ Wpost0, const float* __restrict__ Wpost1, const float* __restrict__ Wpost2,
    const float* __restrict__ Wsc0, const float* __restrict__ Wsc1,
    float* __restrict__ out, int N, int ntiles) {

  __shared__ float lw[10240];            // 8 weight matrices, pair-interleaved (40 KB)
  __shared__ float st[NWAVES * 512];     // per-wave 16x32 staging (8 KB)

  const int tid = threadIdx.x;
  auto fill = [&](const float* __restrict__ W, int K, int base) {
    for (int idx = tid; idx < K * 32; idx += NWAVES * 32) {
      int k = idx >> 5, col = idx & 31;
      lw[base + (k >> 1) * 64 + col * 2 + (k & 1)] = W[idx];
    }
  };
  fill(Wpre0, 64, 0);
  fill(Wpre1, 64, 2048);
  fill(Wpre2, 32, 4096);
  fill(Wpost0, 32, 5120);
  fill(Wpost1, 32, 6144);
  fill(Wpost2, 32, 7168);
  fill(Wsc0, 32, 8192);
  fill(Wsc1, 32, 9216);
  __syncthreads();

  const int wave = tid >> 5;
  const int lane = tid & 31;
  const int tile = blockIdx.x * NWAVES + wave;
  if (tile >= ntiles) return;
  const int n0 = tile * 16;
  float* stw = st + wave * 512;

  // A-fragment row, clamped so partial tail tiles never read out of bounds
  int row = lane & 15;
  if (n0 + row >= N) row = (N - 1) - n0;

  // ---- scalar channel: h0 = gelu(a0 @ Wpre0); o0 = h0@Wpost0 + node0@Wsc0 ----
  {
    v8f h[2] = {v8zero(), v8zero()};
    gemm16<64>(a0 + (size_t)n0 * 64, 64, 1, lw + 0, row, lane, h);
#pragma unroll
    for (int nt = 0; nt < 2; ++nt)
#pragma unroll
      for (int r = 0; r < 8; ++r) h[nt][r] = gelu_tanh(h[nt][r]);
    stage16x32(stw, h, lane);
    asm volatile("s_wait_dscnt 0" ::: "memory");

    v8f o[2] = {v8zero(), v8zero()};
    gemm16<32>(stw, 32, 1, lw + 5120, row, lane, o);                       // h0 @ Wpost0
    gemm16<32>(node0 + (size_t)n0 * 32, 32, 1, lw + 8192, row, lane, o);   // + node0 @ Wsc0
    store_out(out, n0, N, lane, o, /*base*/0, /*stride*/1, /*off*/0);
  }

  // ---- vector channel (1o), per cartesian component i ----
#pragma unroll
  for (int i = 0; i < 3; ++i) {
    v8f h[2] = {v8zero(), v8zero()};
    gemm16<64>(a1 + ((size_t)n0 * 3 + i) * 64, 192, 1, lw + 2048, row, lane, h);
    asm volatile("s_wait_dscnt 0" ::: "memory");
    stage16x32(stw, h, lane);
    asm volatile("s_wait_dscnt 0" ::: "memory");

    v8f o[2] = {v8zero(), v8zero()};
    gemm16<32>(stw, 32, 1, lw + 6144, row, lane, o);                             // h1 @ Wpost1
    gemm16<32>(node1 + (size_t)n0 * 96 + i, 96, 3, lw + 9216, row, lane, o);     // + node1 @ Wsc1
    store_out(out, n0, N, lane, o, /*base*/32, /*stride*/3, /*off*/i);
  }

  // ---- 2e channel, per m component ----
#pragma unroll
  for (int m = 0; m < 5; ++m) {
    v8f h[2] = {v8zero(), v8zero()};
    gemm16<32>(a2 + ((size_t)n0 * 5 + m) * 32, 160, 1, lw + 4096, row, lane, h);
    asm volatile("s_wait_dscnt 0" ::: "memory");
    stage16x32(stw, h, lane);
    asm volatile("s_wait_dscnt 0" ::: "memory");

    v8f o[2] = {v8zero(), v8zero()};
    gemm16<32>(stw, 32, 1, lw + 7168, row, lane, o);                       // h2 @ Wpost2
    store_out(out, n0, N, lane, o, /*base*/128, /*stride*/5, /*off*/m);
  }
}

// ---------------------------------------------------------------------------
extern "C" void kernel_launch(void* const* d_in, const int* in_sizes, int n_in,
                              void* d_out, int out_size, void* d_ws, size_t ws_size,
                              hipStream_t stream) {
  const float* node0 = (const float*)d_in[0];
  const float* node1 = (const float*)d_in[1];
  const float* pos   = (const float*)d_in[2];
  const int* senders = (const int*)d_in[3];
  const int* recvs   = (const int*)d_in[4];
  const float* Wpre0  = (const float*)d_in[5];
  const float* Wpre1  = (const float*)d_in[6];
  const float* Wpre2  = (const float*)d_in[7];
  const float* Wpost0 = (const float*)d_in[8];
  const float* Wpost1 = (const float*)d_in[9];
  const float* Wpost2 = (const float*)d_in[10];
  const float* Wsc0   = (const float*)d_in[11];
  const float* Wsc1   = (const float*)d_in[12];
  float* out = (float*)d_out;

  const int N = in_sizes[0] / 32;
  const int E = in_sizes[3];

  float* a0 = (float*)d_ws;                 // N*64
  float* a1 = a0 + (size_t)N * 64;          // N*192
  float* a2 = a1 + (size_t)N * 192;         // N*160
  const size_t acc_total = (size_t)N * 416;

  zero_kernel<<<2048, 256, 0, stream>>>(a0, acc_total);
  edge_kernel<<<(E + 7) / 8, 256, 0, stream>>>(node0, node1, pos, senders, recvs,
                                               a0, a1, a2, E);
  const int ntiles = (N + 15) / 16;
  node_kernel<<<(ntiles + NWAVES - 1) / NWAVES, NWAVES * 32, 0, stream>>>(
      a0, a1, a2, node0, node1,
      Wpre0, Wpre1, Wpre2, Wpost0, Wpost1, Wpost2, Wsc0, Wsc1,
      out, N, ntiles);
}